// EstimatorCV_69063074120047
// MI455X (gfx1250) — compile-verified
//
#include <hip/hip_runtime.h>
#include <hip/hip_bf16.h>

typedef __attribute__((ext_vector_type(2))) float v2f;
typedef __attribute__((ext_vector_type(8))) float v8f;

#define N_TOT   131072
#define A_DIM   256
#define C_CLS   19
#define C_PAD   32          // padded class count (two 16-row WMMA tiles)
#define WGS     512         // workgroups for the reduction kernel
#define CHUNK   (N_TOT / WGS)   // 256 samples per workgroup

// ---------------------------------------------------------------------------
// Per-class label histogram -> cnt[C_PAD] (float), via LDS then global atomics
// ---------------------------------------------------------------------------
__global__ __launch_bounds__(256) void estcv_cnt_kernel(
    const int* __restrict__ labels, float* __restrict__ cnt, int n) {
  __shared__ float h[C_PAD];
  int t = threadIdx.x;
  if (t < C_PAD) h[t] = 0.0f;
  __syncthreads();
  for (int i = blockIdx.x * blockDim.x + t; i < n; i += gridDim.x * blockDim.x) {
    int l = labels[i] & (C_PAD - 1);
    atomicAdd(&h[l], 1.0f);
  }
  __syncthreads();
  if (t < C_PAD) atomicAdd(&cnt[t], h[t]);
}

// ---------------------------------------------------------------------------
// WMMA segmented reduction: S[c,a] = sum_n onehot[n,c]*F[n,a],
//                           Q[c,a] = sum_n onehot[n,c]*F[n,a]^2
// One wave per 16-column feature slice; K stepped 4 samples at a time with
// V_WMMA_F32_16X16X4_F32 (pure-fp32 matrix pipe, exact accumulation).
// ---------------------------------------------------------------------------
__global__ __launch_bounds__(512, 1) void estcv_wmma_kernel(
    const float* __restrict__ F, const int* __restrict__ L,
    float* __restrict__ wsum, float* __restrict__ wsq) {
  const int lane = threadIdx.x & 31;
  const int wave = threadIdx.x >> 5;      // 0..15 -> feature slice
  const int colb = wave << 4;             // column base of this wave's slice
  const int half = lane >> 4;             // lane half selects samples {0,1} vs {2,3}
  const int l16  = lane & 15;

  const long base = (long)blockIdx.x * CHUNK;

  v8f s0 = {}, s1 = {}, q0 = {}, q1 = {};   // [classes 0-15 | 16-31] x [sum|sumsq]

  for (int k = 0; k < CHUNK; k += 4) {
    const long sI = base + k;
    // wave-uniform label fetches -> scalar loads
    const int lab0 = L[sI + 0];
    const int lab1 = L[sI + 1];
    const int lab2 = L[sI + 2];
    const int lab3 = L[sI + 3];
    const int le = half ? lab2 : lab0;    // label for A-matrix K-even slot
    const int lo = half ? lab3 : lab1;    // label for A-matrix K-odd slot

    // B (4x16 f32): v0 = {K0 | K2}, v1 = {K1 | K3}; N = colb + l16
    const float be = F[(sI + (half ? 2 : 0)) * A_DIM + colb + l16];
    const float bo = F[(sI + (half ? 3 : 1)) * A_DIM + colb + l16];
    v2f b;  b.x  = be;      b.y  = bo;
    v2f bq; bq.x = be * be; bq.y = bo * bo;

    // A (16x4 f32 one-hot): v0 = {K0 | K2}, v1 = {K1 | K3}; M = l16 (+ tile base)
    v2f a0; a0.x = (le == l16)        ? 1.0f : 0.0f;
            a0.y = (lo == l16)        ? 1.0f : 0.0f;
    v2f a1; a1.x = (le == (16 + l16)) ? 1.0f : 0.0f;
            a1.y = (lo == (16 + l16)) ? 1.0f : 0.0f;

    s0 = __builtin_amdgcn_wmma_f32_16x16x4_f32(false, a0, false, b,  (short)0, s0, false, false);
    s1 = __builtin_amdgcn_wmma_f32_16x16x4_f32(false, a1, false, b,  (short)0, s1, false, false);
    q0 = __builtin_amdgcn_wmma_f32_16x16x4_f32(false, a0, false, bq, (short)0, q0, false, false);
    q1 = __builtin_amdgcn_wmma_f32_16x16x4_f32(false, a1, false, bq, (short)0, q1, false, false);
  }

  // C/D layout: VGPR r holds row r (lanes 0-15) / r+8 (lanes 16-31), N = l16
  const int col = colb + l16;
#pragma unroll
  for (int r = 0; r < 8; ++r) {
    const int m = r + (half << 3);
    atomicAdd(&wsum[(m)      * A_DIM + col], s0[r]);
    atomicAdd(&wsum[(m + 16) * A_DIM + col], s1[r]);
    atomicAdd(&wsq [(m)      * A_DIM + col], q0[r]);
    atomicAdd(&wsq [(m + 16) * A_DIM + col], q1[r]);
  }
}

// ---------------------------------------------------------------------------
// Finalize: elementwise EstimatorCV update over C x A, plus amount_new[C].
// d_out = [cov_new (C*A) | ave_new (C*A) | amount_new (C)]
// ---------------------------------------------------------------------------
__global__ __launch_bounds__(256) void estcv_finalize_kernel(
    const float* __restrict__ cnt, const float* __restrict__ wsum,
    const float* __restrict__ wsq, const float* __restrict__ Cov,
    const float* __restrict__ Ave, const float* __restrict__ Amount,
    float* __restrict__ out) {
  const int CA = C_CLS * A_DIM;
  int idx = blockIdx.x * blockDim.x + threadIdx.x;
  if (idx >= CA) return;
  const int c = idx / A_DIM;
  const int a = idx - c * A_DIM;

  const float n   = cnt[c];
  const float s   = wsum[c * A_DIM + a];
  const float sq  = wsq [c * A_DIM + a];
  const float nc  = fmaxf(n, 1.0f);
  const float ave = s / nc;
  const float var = (sq - 2.0f * ave * s + n * ave * ave) / nc;
  const float den = n + Amount[c];
  const float w   = (den > 0.0f) ? (n / den) : 0.0f;
  const float om  = 1.0f - w;
  const float d   = Ave[idx] - ave;

  out[idx]      = Cov[idx] * om + var * w + w * om * d * d;   // cov_new
  out[CA + idx] = Ave[idx] * om + ave * w;                    // ave_new
  if (a == 0) out[2 * CA + c] = Amount[c] + n;                // amount_new
}

// ---------------------------------------------------------------------------
extern "C" void kernel_launch(void* const* d_in, const int* in_sizes, int n_in,
                              void* d_out, int out_size, void* d_ws, size_t ws_size,
                              hipStream_t stream) {
  const float* features = (const float*)d_in[0];
  const int*   labels   = (const int*)  d_in[1];
  const float* Cov      = (const float*)d_in[2];
  const float* Ave      = (const float*)d_in[3];
  const float* Amount   = (const float*)d_in[4];
  float* out = (float*)d_out;

  // workspace: cnt[32] | wsum[32*256] | wsq[32*256]
  float* cnt  = (float*)d_ws;
  float* wsum = cnt + C_PAD;
  float* wsq  = wsum + C_PAD * A_DIM;
  const size_t ws_bytes = (size_t)(C_PAD + 2 * C_PAD * A_DIM) * sizeof(float);
  hipMemsetAsync(d_ws, 0, ws_bytes, stream);

  estcv_cnt_kernel<<<64, 256, 0, stream>>>(labels, cnt, N_TOT);
  estcv_wmma_kernel<<<WGS, 512, 0, stream>>>(features, labels, wsum, wsq);
  estcv_finalize_kernel<<<(C_CLS * A_DIM + 255) / 256, 256, 0, stream>>>(
      cnt, wsum, wsq, Cov, Ave, Amount, out);
}